// MaskRCNN_41446434407127
// MI455X (gfx1250) — compile-verified
//
#include <hip/hip_runtime.h>
#include <stdint.h>

// Problem geometry (fixed by the reference)
#define B_EX 2
#define DD 32
#define HH 96
#define WW 96
#define MTOT (DD * HH * WW)      // 294912 boxes per example
#define NB 32                    // workgroups per example (one WGP each)
#define BS (MTOT / NB)           // 9216 boxes per workgroup slice
#define NT 256                   // threads per workgroup (8 waves of 32)
#define PT (BS / NT)             // 36 boxes per thread
#define KSEL 200                 // boxes kept per example
#define ANCH 8.0f

#define NEGINF (-__builtin_huge_valf())

#if defined(__HIP_DEVICE_COMPILE__) && __has_builtin(__builtin_amdgcn_global_load_async_to_lds_b32)
#define HAVE_ASYNC_LDS 1
#else
#define HAVE_ASYNC_LDS 0
#endif

typedef __attribute__((address_space(1))) int g_int;
typedef __attribute__((address_space(3))) int l_int;

// Order-preserving float -> u32 key (monotonic: bigger float => bigger key)
__device__ __forceinline__ uint32_t enc_f32(float f) {
  uint32_t b = __float_as_uint(f);
  return (b & 0x80000000u) ? ~b : (b | 0x80000000u);
}
__device__ __forceinline__ float dec_f32(uint32_t h) {
  return (h & 0x80000000u) ? __uint_as_float(h ^ 0x80000000u)
                           : __uint_as_float(~h);
}

// Zero the inter-block communication area every launch (graph-replay safe:
// the harness does NOT re-poison d_ws between replays, so we must).
__global__ void nms_init_ws(uint32_t* ws) {
  int t = threadIdx.x + blockIdx.x * blockDim.x;
  if (t < 1024) ws[t] = 0;
}

// Persistent fused NMS kernel: 64 blocks total (32 per example), entire
// per-slice box state lives in LDS (294 KB of the WGP's 320 KB).
__global__ __launch_bounds__(NT, 1)
void nms_persistent(const float* __restrict__ bb,   // [B,6,D,H,W]
                    const float* __restrict__ scg,  // [B,D,H,W]
                    float* __restrict__ out,        // [B,K,7]
                    uint32_t* __restrict__ ws) {
  extern __shared__ float smem[];
  float* loz = smem + 0 * BS;
  float* loy = smem + 1 * BS;
  float* lox = smem + 2 * BS;
  float* hiz = smem + 3 * BS;
  float* hiy = smem + 4 * BS;
  float* hix = smem + 5 * BS;
  float* vol = smem + 6 * BS;
  float* sc  = smem + 7 * BS;

  __shared__ float redv[NT / 32];
  __shared__ int   redi[NT / 32];
  __shared__ float bc[7];   // cloz,cloy,clox,chiz,chiy,chix, score

  const int ex   = blockIdx.x / NB;
  const int blk  = blockIdx.x % NB;
  const int t    = threadIdx.x;
  const int base = blk * BS;

  uint32_t* ctr   = ws;        // ws[0], ws[1]: per-example arrival counters
  uint32_t* slots = ws + 16;   // 8 u32 per (ex,blk): key_lo, key_hi, lo3, hi3

  const float* bex = bb  + (size_t)ex * 6 * MTOT;
  const float* sex = scg + (size_t)ex * MTOT;

  // ---- Phase 0: deparametrize slice into LDS (SoA, conflict-free) --------
  for (int i = 0; i < PT; ++i) {
    int j = t + i * NT;        // stride-1 across lanes: coalesced + no bank conflicts
    int g = base + j;
    int z = g / (HH * WW);
    int r = g - z * (HH * WW);
    int y = r / WW;
    int x = r - y * WW;
    float cz = bex[0 * MTOT + g] * ANCH + ((float)z + 0.5f);
    float cy = bex[1 * MTOT + g] * ANCH + ((float)y + 0.5f);
    float cx = bex[2 * MTOT + g] * ANCH + ((float)x + 0.5f);
    float sz = __expf(bex[3 * MTOT + g]) * ANCH;
    float sy = __expf(bex[4 * MTOT + g]) * ANCH;
    float sx = __expf(bex[5 * MTOT + g]) * ANCH;
    loz[j] = cz - 0.5f * sz;  hiz[j] = cz + 0.5f * sz;
    loy[j] = cy - 0.5f * sy;  hiy[j] = cy + 0.5f * sy;
    lox[j] = cx - 0.5f * sx;  hix[j] = cx + 0.5f * sx;
    vol[j] = sz * sy * sx;
  }

  // Scores: CDNA5 async global->LDS copy (ASYNCcnt path), fallback to VGPR copy.
  for (int i = 0; i < PT; ++i) {
    int j = t + i * NT;
#if HAVE_ASYNC_LDS
    __builtin_amdgcn_global_load_async_to_lds_b32(
        (g_int*)(sex + base + j),
        (l_int*)&sc[j], 0, 0);
#else
    sc[j] = sex[base + j];
#endif
  }
  asm volatile("s_wait_asynccnt 0" ::: "memory");
  __syncthreads();

  // ---- Main loop: 200 fused suppress(prev) + argmax(next) rounds ---------
  // Sentinel "previous box" that overlaps nothing (round 0 = plain argmax).
  float cloz = 1e30f, cloy = 1e30f, clox = 1e30f;
  float chiz = 1e30f, chiy = 1e30f, chix = 1e30f;
  float cvol = 0.0f;

  for (int k = 0; k < KSEL; ++k) {
    float bv = NEGINF;
    int   bj = 0;
    for (int i = 0; i < PT; ++i) {
      int j = t + i * NT;
      float s = sc[j];
      if (s != NEGINF) {
        float oz = fminf(chiz, hiz[j]) - fmaxf(cloz, loz[j]);
        float oy = fminf(chiy, hiy[j]) - fmaxf(cloy, loy[j]);
        float ox = fminf(chix, hix[j]) - fmaxf(clox, lox[j]);
        if (oz > 0.0f && oy > 0.0f && ox > 0.0f) {
          float inter = oz * oy * ox;
          float uni   = vol[j] + cvol - inter;
          if (inter + inter >= uni) {  // IoU >= 0.5
            sc[j] = NEGINF;
            s = NEGINF;
          }
        }
        if (s > bv) { bv = s; bj = j; }
      }
    }

    // Wave32 butterfly argmax
    for (int o = 16; o > 0; o >>= 1) {
      float ov = __shfl_xor(bv, o, 32);
      int   oj = __shfl_xor(bj, o, 32);
      if (ov > bv) { bv = ov; bj = oj; }
    }
    if ((t & 31) == 0) { redv[t >> 5] = bv; redi[t >> 5] = bj; }
    __syncthreads();

    // Block winner -> global slot, then agent-scope arrival barrier
    if (t == 0) {
      float v = redv[0];
      int   j = redi[0];
      for (int w = 1; w < NT / 32; ++w)
        if (redv[w] > v) { v = redv[w]; j = redi[w]; }
      uint32_t* sl = slots + (size_t)(ex * NB + blk) * 8;
      sl[0] = (uint32_t)(base + j);
      sl[1] = enc_f32(v);
      sl[2] = __float_as_uint(loz[j]);
      sl[3] = __float_as_uint(loy[j]);
      sl[4] = __float_as_uint(lox[j]);
      sl[5] = __float_as_uint(hiz[j]);
      sl[6] = __float_as_uint(hiy[j]);
      sl[7] = __float_as_uint(hix[j]);
      __threadfence();
      __hip_atomic_fetch_add(&ctr[ex], 1u, __ATOMIC_RELEASE,
                             __HIP_MEMORY_SCOPE_AGENT);
      uint32_t target = (uint32_t)NB * (uint32_t)(k + 1);  // monotonic: no sense reversal
      while (__hip_atomic_load(&ctr[ex], __ATOMIC_ACQUIRE,
                               __HIP_MEMORY_SCOPE_AGENT) < target) {
        __builtin_amdgcn_s_sleep(8);
      }
    }
    __syncthreads();

    // Wave 0: max-reduce the 32 per-block keys, broadcast winning box
    if (t < 32) {
      const uint32_t* sl = slots + (size_t)(ex * NB + t) * 8;
      uint32_t klo = sl[0];
      uint32_t khi = sl[1];
      int src = t;
      for (int o = 16; o > 0; o >>= 1) {
        uint32_t ohi = __shfl_xor(khi, o, 32);
        uint32_t olo = __shfl_xor(klo, o, 32);
        int osrc     = __shfl_xor(src, o, 32);
        if (ohi > khi || (ohi == khi && olo > klo)) {
          khi = ohi; klo = olo; src = osrc;
        }
      }
      if (t == 0) {
        const uint32_t* wsl = slots + (size_t)(ex * NB + src) * 8;
        bc[0] = __uint_as_float(wsl[2]);
        bc[1] = __uint_as_float(wsl[3]);
        bc[2] = __uint_as_float(wsl[4]);
        bc[3] = __uint_as_float(wsl[5]);
        bc[4] = __uint_as_float(wsl[6]);
        bc[5] = __uint_as_float(wsl[7]);
        bc[6] = dec_f32(khi);
      }
    }
    __syncthreads();

    cloz = bc[0]; cloy = bc[1]; clox = bc[2];
    chiz = bc[3]; chiy = bc[4]; chix = bc[5];
    cvol = (chiz - cloz) * (chiy - cloy) * (chix - clox);

    if (blk == 0 && t == 0) {
      float* o = out + (size_t)(ex * KSEL + k) * 7;
      o[0] = bc[6];                      // score
      o[1] = 0.5f * (cloz + chiz);       // center z
      o[2] = 0.5f * (cloy + chiy);       // center y
      o[3] = 0.5f * (clox + chix);       // center x
      o[4] = chiz - cloz;                // size d
      o[5] = chiy - cloy;                // size h
      o[6] = chix - clox;                // size w
    }
    // bc is only rewritten after next round's first __syncthreads, so the
    // register copies above are race-free.
  }
}

extern "C" void kernel_launch(void* const* d_in, const int* in_sizes, int n_in,
                              void* d_out, int out_size, void* d_ws, size_t ws_size,
                              hipStream_t stream) {
  const float* bboxes = (const float*)d_in[0];  // [2,6,32,96,96] f32
  const float* scores = (const float*)d_in[1];  // [2,32,96,96]  f32
  float* out = (float*)d_out;                   // [2,200,7]     f32
  uint32_t* ws = (uint32_t*)d_ws;

  // Reset barrier counters + slots every launch (graph replays re-run this).
  nms_init_ws<<<4, 256, 0, stream>>>(ws);

  // 64 persistent blocks, 294912 B dynamic LDS each (one block per WGP).
  nms_persistent<<<B_EX * NB, NT, 8 * BS * sizeof(float), stream>>>(
      bboxes, scores, out, ws);
}